// EdgewiseEnergySum_46883863003658
// MI455X (gfx1250) — compile-verified
//
#include <hip/hip_runtime.h>

// Edgewise energy scatter-sum for MI455X (gfx1250, wave32).
// out[center[e]] += eng[e] * scales[type[center[e]], type[neighbor[e]]] * FACTOR
//
// Memory-bound: ~77 MB of read-once edge streams + random 400KB atom_type
// gathers + 6.4M float atomic adds into 400KB of output (L2-side atomics).
// Strategy: 128-bit non-temporal streaming loads for the edge arrays (keep
// the hot atom_type / out working set cache-resident), LDS-resident
// pre-scaled 8x8 table, no-return global_atomic_add_f32 scatter.

typedef int   __attribute__((ext_vector_type(4))) vi4;
typedef float __attribute__((ext_vector_type(4))) vf4;

#define NUM_TYPES 8
static constexpr float kFactor = 0.125f; // 1/sqrt(64)

__global__ void EdgewiseZero_kernel(float* __restrict__ out, int n) {
    int i = blockIdx.x * blockDim.x + threadIdx.x;
    if (i < n) out[i] = 0.0f;
}

__global__ __launch_bounds__(256) void EdgewiseEnergySum_46883863003658_kernel(
    const int*   __restrict__ ecen,    // [E] edge_index row 0 (centers)
    const int*   __restrict__ enei,    // [E] edge_index row 1 (neighbors)
    const float* __restrict__ eng,     // [E] edge energies
    const int*   __restrict__ atype,   // [N] atom types (0..7)
    const float* __restrict__ scales,  // [8,8] per-edge scale table
    float*       __restrict__ out,     // [N] accumulated energies (pre-zeroed)
    int n_edges)
{
    // Pre-scaled lookup table in LDS: folds FACTOR into the table once per block.
    __shared__ float s_scale[NUM_TYPES * NUM_TYPES];
    if (threadIdx.x < NUM_TYPES * NUM_TYPES)
        s_scale[threadIdx.x] = scales[threadIdx.x] * kFactor;
    __syncthreads();

    const int base = (blockIdx.x * blockDim.x + threadIdx.x) * 4;

    if (base + 4 <= n_edges) {
        // 128-bit non-temporal streaming loads (read-once data; TH=NT keeps
        // atom_type/out resident in cache instead of this 77MB stream).
        vi4 c  = __builtin_nontemporal_load((const vi4*)(ecen + base));
        vi4 nb = __builtin_nontemporal_load((const vi4*)(enei + base));
        vf4 e  = __builtin_nontemporal_load((const vf4*)(eng  + base));

        // Gather atom types (random access, cache-friendly working set).
        int tc0 = atype[c.x],  tc1 = atype[c.y],  tc2 = atype[c.z],  tc3 = atype[c.w];
        int tn0 = atype[nb.x], tn1 = atype[nb.y], tn2 = atype[nb.z], tn3 = atype[nb.w];

        float v0 = e.x * s_scale[tc0 * NUM_TYPES + tn0];
        float v1 = e.y * s_scale[tc1 * NUM_TYPES + tn1];
        float v2 = e.z * s_scale[tc2 * NUM_TYPES + tn2];
        float v3 = e.w * s_scale[tc3 * NUM_TYPES + tn3];

        // No-return, relaxed, device-scope fp32 atomic adds -> global_atomic_add_f32.
        (void)__hip_atomic_fetch_add(&out[c.x], v0, __ATOMIC_RELAXED, __HIP_MEMORY_SCOPE_AGENT);
        (void)__hip_atomic_fetch_add(&out[c.y], v1, __ATOMIC_RELAXED, __HIP_MEMORY_SCOPE_AGENT);
        (void)__hip_atomic_fetch_add(&out[c.z], v2, __ATOMIC_RELAXED, __HIP_MEMORY_SCOPE_AGENT);
        (void)__hip_atomic_fetch_add(&out[c.w], v3, __ATOMIC_RELAXED, __HIP_MEMORY_SCOPE_AGENT);
    } else {
        // Tail (E % 4 != 0 safety; E = 6.4M is divisible by 4 but stay general).
        for (int i = base; i < n_edges; ++i) {
            int ci = ecen[i];
            int ni = enei[i];
            float v = eng[i] * s_scale[atype[ci] * NUM_TYPES + atype[ni]];
            (void)__hip_atomic_fetch_add(&out[ci], v, __ATOMIC_RELAXED, __HIP_MEMORY_SCOPE_AGENT);
        }
    }
}

extern "C" void kernel_launch(void* const* d_in, const int* in_sizes, int n_in,
                              void* d_out, int out_size, void* d_ws, size_t ws_size,
                              hipStream_t stream) {
    // setup_inputs() order: edge_index [2,E] i32, edge_eng [E,1] f32,
    // atom_type [N,1] i32, per_edge_scales [8,8] f32.
    const int n_edges = in_sizes[0] / 2;

    const int*   ecen   = (const int*)d_in[0];          // row 0: centers
    const int*   enei   = ecen + n_edges;               // row 1: neighbors
    const float* eng    = (const float*)d_in[1];
    const int*   atype  = (const int*)d_in[2];
    const float* scales = (const float*)d_in[3];
    float*       out    = (float*)d_out;

    // 1) Zero the output (harness poisons d_out; atomics accumulate onto 0).
    {
        const int threads = 256;
        const int blocks = (out_size + threads - 1) / threads;
        EdgewiseZero_kernel<<<blocks, threads, 0, stream>>>(out, out_size);
    }

    // 2) Scatter-accumulate all edges, 4 edges per thread.
    {
        const int threads = 256;                 // 8 wave32 waves per block
        const int per_block = threads * 4;
        const int blocks = (n_edges + per_block - 1) / per_block;
        EdgewiseEnergySum_46883863003658_kernel<<<blocks, threads, 0, stream>>>(
            ecen, enei, eng, atype, scales, out, n_edges);
    }
}